// MZ_model_FNN_86663850099032
// MI455X (gfx1250) — compile-verified
//
#include <hip/hip_runtime.h>
#include <hip/hip_bf16.h>

typedef __attribute__((ext_vector_type(2))) float v2f;
typedef __attribute__((ext_vector_type(8))) float v8f;

#define T_DIM 512
#define B_DIM 256
#define O_DIM 64
#define I_DIM 512
#define H_DIM 1024
#define F_DIM 640            // 2*O + I
#define M_TILE 32
#define LDS_STRIDE 1028      // 1024 + 4 pad (bank-conflict-free A frag reads)
#define IN_STRIDE 68         // 64 + 4 pad

// ---------------------------------------------------------------------------
// Weight transpose: src (R x C) -> dst (C x R). R, C multiples of 32.
// ---------------------------------------------------------------------------
__global__ __launch_bounds__(256) void k_transpose(const float* __restrict__ src,
                                                   float* __restrict__ dst,
                                                   int R, int C) {
    __shared__ float tile[32][33];
    const int tx = threadIdx.x & 31;
    const int g  = threadIdx.x >> 5;          // 0..7, each handles 4 rows
    const int c0 = blockIdx.x * 32;
    const int r0 = blockIdx.y * 32;
#pragma unroll
    for (int i = 0; i < 4; ++i) {
        int r = g * 4 + i;
        tile[r][tx] = src[(size_t)(r0 + r) * C + c0 + tx];
    }
    __syncthreads();
#pragma unroll
    for (int i = 0; i < 4; ++i) {
        int c = g * 4 + i;
        dst[(size_t)(c0 + c) * R + r0 + tx] = tile[tx][c];
    }
}

// ---------------------------------------------------------------------------
// Fragment helpers for V_WMMA_F32_16X16X4_F32 (wave32).
//   A (16x4): lane m = lane&15, kb = (lane>>4)*2 -> {A[m][k+kb], A[m][k+kb+1]}
//   B (4x16): lane n = lane&15, kb = (lane>>4)*2 -> {B[k+kb][n], B[k+kb+1][n]}
//   C (16x16): reg j -> row j + 8*(lane>=16), col lane&15
// ---------------------------------------------------------------------------
__device__ __forceinline__ v2f afrag_lds(const float* tile, int stride, int m,
                                         int k, int kb) {
    v2f a;
    a.x = tile[m * stride + k + kb];
    a.y = tile[m * stride + k + kb + 1];
    return a;
}

__device__ __forceinline__ v2f bfrag_glb(const float* __restrict__ WT, int ldw,
                                         int k, int kb, int col) {
    const float* p = WT + (size_t)(k + kb) * ldw + col;
    v2f b;
    b.x = p[0];
    b.y = p[ldw];
    return b;
}

#define WMMA_F32(a, b, c) \
    __builtin_amdgcn_wmma_f32_16x16x4_f32(false, (a), false, (b), (short)0, (c), false, false)

// ---------------------------------------------------------------------------
// Fused 3-layer MLP + drive epilogue.
// Block: 256 threads (8 waves). Tile: 32 rows through all layers.
// Waves split H into 8 chunks of 128 cols; each wave: 2 m-tiles x 8 n-tiles.
// ---------------------------------------------------------------------------
__global__ __launch_bounds__(256, 1)
void k_mlp(const float* __restrict__ u,
           const float* __restrict__ W0T, const float* __restrict__ b0,
           const float* __restrict__ W1T, const float* __restrict__ b1,
           const float* __restrict__ W2T, const float* __restrict__ b2,
           const float* __restrict__ bvec,
           float* __restrict__ out) {
    __shared__ float sh0[M_TILE * LDS_STRIDE];      // h0 tile (relu out, layer0)
    __shared__ float sh1[M_TILE * LDS_STRIDE];      // h1 tile (relu out, layer1)
    __shared__ float s_in[2][M_TILE * IN_STRIDE];   // double-buffered input panel

    const int tid  = threadIdx.x;
    const int wv   = tid >> 5;
    const int lane = tid & 31;
    const int nsel = lane & 15;
    const int kb   = (lane >> 4) << 1;              // 0 or 2
    const int mhi  = (lane >> 4) << 3;              // 0 or 8 (C row offset)
    const size_t row_base = (size_t)blockIdx.x * M_TILE;

    const v8f vzero = {0.f, 0.f, 0.f, 0.f, 0.f, 0.f, 0.f, 0.f};
    v8f acc[16];

    // ================= Layer 0: h0 = relu(net_in @ W0T + b0), K = 512 ======
#pragma unroll
    for (int i = 0; i < 16; ++i) acc[i] = vzero;

    // panel loader: 32 rows x 64 cols of net_in, coalesced float4
    const int e0 = tid, e1 = tid + 256;             // 512 float4 per panel
    float4 pr0, pr1;
    {
        const float* base = u + row_base * F_DIM + 2 * O_DIM;
        pr0 = *(const float4*)(base + (size_t)(e0 >> 4) * F_DIM + ((e0 & 15) << 2));
        pr1 = *(const float4*)(base + (size_t)(e1 >> 4) * F_DIM + ((e1 & 15) << 2));
        *(float4*)&s_in[0][(e0 >> 4) * IN_STRIDE + ((e0 & 15) << 2)] = pr0;
        *(float4*)&s_in[0][(e1 >> 4) * IN_STRIDE + ((e1 & 15) << 2)] = pr1;
    }
    __syncthreads();

    const int ncb = wv * 128 + nsel;                // this wave's base column

    for (int p = 0; p < 8; ++p) {
        if (p < 7) {
            const float* base = u + row_base * F_DIM + 2 * O_DIM + ((p + 1) << 6);
            pr0 = *(const float4*)(base + (size_t)(e0 >> 4) * F_DIM + ((e0 & 15) << 2));
            pr1 = *(const float4*)(base + (size_t)(e1 >> 4) * F_DIM + ((e1 & 15) << 2));
        }
        const float* tileA = s_in[p & 1];
#pragma unroll 4
        for (int kk = 0; kk < 64; kk += 4) {
            const int kg = (p << 6) + kk;
            v2f a0 = afrag_lds(tileA, IN_STRIDE, nsel, kk, kb);
            v2f a1 = afrag_lds(tileA, IN_STRIDE, 16 + nsel, kk, kb);
#pragma unroll
            for (int nt = 0; nt < 8; ++nt) {
                v2f bf = bfrag_glb(W0T, H_DIM, kg, kb, ncb + nt * 16);
                acc[nt]     = WMMA_F32(a0, bf, acc[nt]);
                acc[8 + nt] = WMMA_F32(a1, bf, acc[8 + nt]);
            }
        }
        if (p < 7) {
            __syncthreads();
            *(float4*)&s_in[(p + 1) & 1][(e0 >> 4) * IN_STRIDE + ((e0 & 15) << 2)] = pr0;
            *(float4*)&s_in[(p + 1) & 1][(e1 >> 4) * IN_STRIDE + ((e1 & 15) << 2)] = pr1;
            __syncthreads();
        }
    }

    // bias + relu -> sh0
#pragma unroll
    for (int nt = 0; nt < 8; ++nt) {
        const int col = ncb + nt * 16;
        const float bb = b0[col];
#pragma unroll
        for (int j = 0; j < 8; ++j) {
            const int m = j + mhi;
            float v0 = acc[nt][j] + bb;
            float v1 = acc[8 + nt][j] + bb;
            sh0[m * LDS_STRIDE + col]        = v0 > 0.f ? v0 : 0.f;
            sh0[(16 + m) * LDS_STRIDE + col] = v1 > 0.f ? v1 : 0.f;
        }
    }
    __syncthreads();

    // ================= Layer 1: h1 = relu(h0 @ W1T + b1), K = 1024 =========
#pragma unroll
    for (int i = 0; i < 16; ++i) acc[i] = vzero;
#pragma unroll 2
    for (int k = 0; k < H_DIM; k += 4) {
        v2f a0 = afrag_lds(sh0, LDS_STRIDE, nsel, k, kb);
        v2f a1 = afrag_lds(sh0, LDS_STRIDE, 16 + nsel, k, kb);
#pragma unroll
        for (int nt = 0; nt < 8; ++nt) {
            v2f bf = bfrag_glb(W1T, H_DIM, k, kb, ncb + nt * 16);
            acc[nt]     = WMMA_F32(a0, bf, acc[nt]);
            acc[8 + nt] = WMMA_F32(a1, bf, acc[8 + nt]);
        }
    }
#pragma unroll
    for (int nt = 0; nt < 8; ++nt) {
        const int col = ncb + nt * 16;
        const float bb = b1[col];
#pragma unroll
        for (int j = 0; j < 8; ++j) {
            const int m = j + mhi;
            float v0 = acc[nt][j] + bb;
            float v1 = acc[8 + nt][j] + bb;
            sh1[m * LDS_STRIDE + col]        = v0 > 0.f ? v0 : 0.f;
            sh1[(16 + m) * LDS_STRIDE + col] = v1 > 0.f ? v1 : 0.f;
        }
    }
    __syncthreads();

    // ================= Layer 2: net_out = h1 @ W2T + b2, N = 64 ============
    // 8 tiles total (2 m x 4 n): one 16x16 tile per wave.
    const int mt2 = wv >> 2;          // 0..1
    const int nt2 = wv & 3;           // 0..3
    v8f c2 = vzero;
#pragma unroll 4
    for (int k = 0; k < H_DIM; k += 4) {
        v2f a  = afrag_lds(sh1, LDS_STRIDE, mt2 * 16 + nsel, k, kb);
        v2f bf = bfrag_glb(W2T, O_DIM, k, kb, nt2 * 16 + nsel);
        c2 = WMMA_F32(a, bf, c2);
    }
    {
        const int col = nt2 * 16 + nsel;
        const float bb2 = b2[col];
        const float bsc = bvec[col];
#pragma unroll
        for (int j = 0; j < 8; ++j) {
            const int m = mt2 * 16 + j + mhi;
            const size_t r = row_base + m;
            float drive = bsc * u[r * F_DIM + O_DIM + col] + (c2[j] + bb2);
            // drive for step t (=r/B) lands in u_y slot t+1; scan overwrites it
            out[(r + B_DIM) * O_DIM + col] = drive;
        }
    }
}

// ---------------------------------------------------------------------------
// y0 copy + comp_pen
// ---------------------------------------------------------------------------
__global__ __launch_bounds__(256) void k_init(const float* __restrict__ u,
                                              const float* __restrict__ bvec,
                                              float* __restrict__ out) {
    const int g = blockIdx.x * 256 + threadIdx.x;
    if (g < B_DIM * O_DIM) {
        const int b = g >> 6, o = g & 63;
        out[g] = u[(size_t)b * F_DIM + o];
    }
    if (g < O_DIM) {
        float bb = bvec[g];
        float r = bb > 0.f ? bb : 0.f;
        out[(size_t)(T_DIM + 1) * B_DIM * O_DIM + g] = r * r;
    }
}

// ---------------------------------------------------------------------------
// Sequential scan: y_t = y_{t-1} @ Wa^T + drive_{t-1}, in place over d_out.
// 64 blocks x 256 threads; block handles 4 batch rows; Wa row in registers.
// ---------------------------------------------------------------------------
__global__ __launch_bounds__(256, 1)
void k_scan(const float* __restrict__ Wa, float* __restrict__ out) {
    __shared__ float ylds[4][64];
    const int o    = threadIdx.x & 63;
    const int rloc = threadIdx.x >> 6;                 // 0..3
    const int brow = blockIdx.x * 4 + rloc;

    float w[64];
#pragma unroll
    for (int k = 0; k < 64; ++k) w[k] = Wa[o * 64 + k];   // Wa[o][k] (y @ Wa^T)

    const size_t idx0   = (size_t)brow * O_DIM + o;
    const size_t stride = (size_t)B_DIM * O_DIM;          // 16384

    ylds[rloc][o] = out[idx0];                            // y0
    __syncthreads();

    float dval = out[idx0 + stride];                      // prefetch drive_0
    for (int t = 1; t <= T_DIM; ++t) {
        float accv = dval;
#pragma unroll
        for (int k = 0; k < 64; ++k) accv += w[k] * ylds[rloc][k];
        if (t < T_DIM) dval = out[idx0 + (size_t)(t + 1) * stride]; // prefetch
        __syncthreads();
        ylds[rloc][o] = accv;
        out[idx0 + (size_t)t * stride] = accv;
        __syncthreads();
    }
}

// ---------------------------------------------------------------------------
extern "C" void kernel_launch(void* const* d_in, const int* in_sizes, int n_in,
                              void* d_out, int out_size, void* d_ws, size_t ws_size,
                              hipStream_t stream) {
    (void)in_sizes; (void)n_in; (void)out_size; (void)ws_size;
    const float* u   = (const float*)d_in[0];
    const float* Wa  = (const float*)d_in[1];
    const float* bv  = (const float*)d_in[2];
    const float* W0  = (const float*)d_in[3];
    const float* b0  = (const float*)d_in[4];
    const float* W1  = (const float*)d_in[5];
    const float* b1  = (const float*)d_in[6];
    const float* W2  = (const float*)d_in[7];
    const float* b2  = (const float*)d_in[8];
    float* out = (float*)d_out;

    float* W0T = (float*)d_ws;                 // (512 x 1024)
    float* W1T = W0T + 512 * 1024;             // (1024 x 1024)
    float* W2T = W1T + 1024 * 1024;            // (1024 x 64)

    k_transpose<<<dim3(16, 32), 256, 0, stream>>>(W0, W0T, 1024, 512);
    k_transpose<<<dim3(32, 32), 256, 0, stream>>>(W1, W1T, 1024, 1024);
    k_transpose<<<dim3(32, 2),  256, 0, stream>>>(W2, W2T, 64, 1024);
    k_init<<<64, 256, 0, stream>>>(u, bv, out);
    k_mlp<<<(T_DIM * B_DIM) / M_TILE, 256, 0, stream>>>(u, W0T, b0, W1T, b1,
                                                        W2T, b2, bv, out);
    k_scan<<<B_DIM / 4, 256, 0, stream>>>(Wa, out);
}